// MultiScaleRetention_55748675502575
// MI455X (gfx1250) — compile-verified
//
#include <hip/hip_runtime.h>
#include <hip/hip_bf16.h>

// ---------------------------------------------------------------------------
// MultiScaleRetention forward for MI455X (gfx1250), fp32 WMMA pipeline.
// B=2 T=2048 E=1024 H=8 dk=128 dv=256 V=2048 NQKV=4096
// LDS layouts are K-pair-interleaved so every WMMA fragment is one b64 load.
// ---------------------------------------------------------------------------

typedef __attribute__((ext_vector_type(2))) float v2f;
typedef __attribute__((ext_vector_type(8))) float v8f;

#define CB 2
#define CT 2048
#define CE 1024
#define CH 8
#define CDK 128
#define CDV 256
#define CV 2048
#define NQKV 4096
#define BT (CB * CT)

static __device__ __forceinline__ v8f wmma4(v2f a, v2f b, v8f c) {
  // D = A(16x4,f32) * B(4x16,f32) + C(16x16,f32)
  return __builtin_amdgcn_wmma_f32_16x16x4_f32(
      /*neg_a=*/false, a, /*neg_b=*/false, b,
      /*c_mod=*/(short)0, c, /*reuse_a=*/false, /*reuse_b=*/false);
}

static __device__ __forceinline__ v8f vzero8() {
  v8f z;
#pragma unroll
  for (int i = 0; i < 8; ++i) z[i] = 0.0f;
  return z;
}

static __device__ __forceinline__ v2f ldsv2(const float* p) {
  return *reinterpret_cast<const v2f*>(p);
}

// ---------------------------------------------------------------------------
// Generic C = A[MxK] @ B[KxN] + bias, fp32 WMMA.
// Workgroup: 128 threads = 4 waves. Tile 128x64; wave w -> rows
// [w*32, w*32+32) (2 m-subtiles x 4 n-tiles = 8 accumulators).
// K staged in LDS in chunks of 32; B tile stored K-pair-interleaved.
// ---------------------------------------------------------------------------
__global__ __launch_bounds__(128) void gemm_wmma_f32(
    const float* __restrict__ A, const float* __restrict__ Bm,
    const float* __restrict__ bias, float* __restrict__ C,
    int M, int N, int K) {
  __shared__ float as[128][34];     // row-major; A frag = contiguous b64
  __shared__ float bs[16][66][2];   // [kpair][n][0/1]; B frag = one b64

  const int tid = threadIdx.x;
  const int lane = tid & 31;
  const int w = tid >> 5;
  const int hi = lane >> 4;   // half-wave: K pair +1 / M row +8 (ISA layout)
  const int ln = lane & 15;
  const int n0 = blockIdx.x * 64;
  const int m0 = blockIdx.y * 128;

  v8f acc[2][4];
#pragma unroll
  for (int mt = 0; mt < 2; ++mt)
#pragma unroll
    for (int nt = 0; nt < 4; ++nt) acc[mt][nt] = vzero8();

  for (int k0 = 0; k0 < K; k0 += 32) {
#pragma unroll
    for (int e = tid; e < 128 * 32; e += 128) {
      int r = e >> 5, c = e & 31;
      as[r][c] = A[(size_t)(m0 + r) * K + (k0 + c)];
    }
#pragma unroll
    for (int e = tid; e < 32 * 64; e += 128) {
      int r = e >> 6, c = e & 63;
      bs[r >> 1][c][r & 1] = Bm[(size_t)(k0 + r) * N + (n0 + c)];
    }
    __syncthreads();

    if (k0 + 32 < K) {
      // CDNA5 speculative prefetch of the next K-chunk (global_prefetch_b8)
      __builtin_prefetch(&A[(size_t)(m0 + tid) * K + (k0 + 32)], 0, 1);
      __builtin_prefetch(&Bm[(size_t)(k0 + 32 + (tid & 31)) * N + n0], 0, 1);
    }

#pragma unroll
    for (int kk = 0; kk < 32; kk += 4) {
      const int ak = kk + hi * 2;          // A column pair start
      const int kp = (kk >> 1) + hi;       // interleaved K-pair index
      v2f a0 = ldsv2(&as[w * 32 + ln][ak]);
      v2f a1 = ldsv2(&as[w * 32 + 16 + ln][ak]);
#pragma unroll
      for (int nt = 0; nt < 4; ++nt) {
        v2f b = ldsv2(&bs[kp][nt * 16 + ln][0]);
        acc[0][nt] = wmma4(a0, b, acc[0][nt]);
        acc[1][nt] = wmma4(a1, b, acc[1][nt]);
      }
    }
    __syncthreads();
  }

#pragma unroll
  for (int mt = 0; mt < 2; ++mt)
#pragma unroll
    for (int nt = 0; nt < 4; ++nt)
#pragma unroll
      for (int vv = 0; vv < 8; ++vv) {
        int row = m0 + w * 32 + mt * 16 + vv + hi * 8;
        int col = n0 + nt * 16 + ln;
        C[(size_t)row * N + col] = acc[mt][nt][vv] + bias[col];
      }
}

// ---------------------------------------------------------------------------
// xPos rotary on q (upscale) and k (downscale), in place on the qkv buffer.
// Grid: BT blocks x 512 threads (one thread per rotary pair).
// ---------------------------------------------------------------------------
__global__ __launch_bounds__(512) void xpos_kernel(float* __restrict__ qkv) {
  const int row = blockIdx.x;        // 0..BT-1
  const int j = threadIdx.x;         // 0..511 (pair index)
  const int t = row & (CT - 1);

  const float D = (float)CE;
  const float base = (2.0f * (float)j + 0.4f * D) / (1.4f * D);
  const float sc = __powf(base, (float)t / 512.0f);
  const float invf = __powf(10000.0f, -(float)j / 512.0f);
  float sn, cn;
  __sincosf((float)t * invf, &sn, &cn);

  size_t off = (size_t)row * NQKV + 2 * j;
  // q: scale = sc
  {
    float x0 = qkv[off], x1 = qkv[off + 1];
    float c = cn * sc, s = sn * sc;
    qkv[off]     = x0 * c - x1 * s;
    qkv[off + 1] = x1 * c + x0 * s;
  }
  // k: scale = 1/sc
  {
    float isc = 1.0f / sc;
    float x0 = qkv[off + CE], x1 = qkv[off + CE + 1];
    float c = cn * isc, s = sn * isc;
    qkv[off + CE]     = x0 * c - x1 * s;
    qkv[off + CE + 1] = x1 * c + x0 * s;
  }
}

// ---------------------------------------------------------------------------
// Retention: for one (b,h) and a 64-row query block, loop causal key blocks
// of 16: S = QK^T (WMMA), analytic decay mask in-fragment, restage S via LDS,
// out += S @ V (WMMA). Row normalization 1/sqrt(sum gamma^(i-j)) applied in
// the epilogue (closed form). Writes retention_out in [B,T,V] layout.
// Grid: (T/64, B*H); block 128 threads = 4 waves; each wave owns 16 rows.
// ---------------------------------------------------------------------------
__global__ __launch_bounds__(128) void retention_kernel(
    const float* __restrict__ qkv, float* __restrict__ ro) {
  __shared__ float q_s[64][132];     // row-major: A frags contiguous
  __shared__ float k_s[16][132];     // row-major: S-phase B frags contiguous
  __shared__ float v_s[8][258][2];   // K-pair-interleaved over j
  __shared__ float s_s[4][16][20];   // per-wave S tile; A frags contiguous

  const int iblk = blockIdx.x;
  const int bh = blockIdx.y;
  const int b = bh >> 3, h = bh & 7;
  const int i0 = iblk * 64;
  const int rowbase = b * CT;

  const int tid = threadIdx.x;
  const int lane = tid & 31;
  const int w = tid >> 5;
  const int hi = lane >> 4;
  const int ln = lane & 15;

  const float se = logf(1.0f / 32.0f), ee = logf(1.0f / 512.0f);
  const float g = 1.0f - __expf(se + (ee - se) * ((float)h / 7.0f));
  const float logg = logf(g);
  const float rsdk = rsqrtf((float)CDK);

  // Stage Q tile (64 x 128) once.
#pragma unroll
  for (int e = tid; e < 64 * 128; e += 128) {
    int r = e >> 7, c = e & 127;
    q_s[r][c] = qkv[(size_t)(rowbase + i0 + r) * NQKV + h * CDK + c];
  }

  v8f acc[16];
#pragma unroll
  for (int i = 0; i < 16; ++i) acc[i] = vzero8();

  for (int j0 = 0; j0 < i0 + 64; j0 += 16) {
    __syncthreads();
#pragma unroll
    for (int e = tid; e < 16 * 128; e += 128) {
      int r = e >> 7, c = e & 127;
      k_s[r][c] = qkv[(size_t)(rowbase + j0 + r) * NQKV + CE + h * CDK + c];
    }
#pragma unroll
    for (int e = tid; e < 16 * 256; e += 128) {
      int r = e >> 8, c = e & 255;
      v_s[r >> 1][c][r & 1] =
          qkv[(size_t)(rowbase + j0 + r) * NQKV + 2 * CE + h * CDV + c];
    }
    __syncthreads();

    // S = Q_block(16x128) @ K_block(16x128)^T  -> 32 WMMA (K=4 each)
    v8f sacc = vzero8();
#pragma unroll
    for (int kk = 0; kk < CDK; kk += 4) {
      const int ak = kk + hi * 2;
      v2f a = ldsv2(&q_s[w * 16 + ln][ak]);
      v2f bf = ldsv2(&k_s[ln][ak]);   // B[kkk][n] = K[j0+n][kk+kkk]
      sacc = wmma4(a, bf, sacc);
    }

    // Decay mask (causal) * dk^-0.5, applied in C-fragment layout.
#pragma unroll
    for (int vv = 0; vv < 8; ++vv) {
      int i = i0 + w * 16 + vv + hi * 8;
      int j = j0 + ln;
      float m = (i >= j) ? __expf(logg * (float)(i - j)) * rsdk : 0.0f;
      sacc[vv] *= m;
    }

    // Restage S (C-layout) -> LDS -> A-fragment layout. Same-wave LDS RAW is
    // in-order (DS ops in order per wave).
#pragma unroll
    for (int vv = 0; vv < 8; ++vv) s_s[w][vv + hi * 8][ln] = sacc[vv];

    // out += S(16x16) @ V_block(16x256) -> 16 n-tiles x 4 k-steps = 64 WMMA
#pragma unroll
    for (int ks = 0; ks < 16; ks += 4) {
      const int ak = ks + hi * 2;
      const int kp = (ks >> 1) + hi;
      v2f a = ldsv2(&s_s[w][ln][ak]);
#pragma unroll
      for (int nt = 0; nt < 16; ++nt) {
        v2f bf = ldsv2(&v_s[kp][nt * 16 + ln][0]);
        acc[nt] = wmma4(a, bf, acc[nt]);
      }
    }
  }

  // Epilogue: per-row normalization rsqrt(sum_{j<=i} gamma^(i-j)).
  const float inv1mg = 1.0f / (1.0f - g);
#pragma unroll
  for (int nt = 0; nt < 16; ++nt)
#pragma unroll
    for (int vv = 0; vv < 8; ++vv) {
      int i = i0 + w * 16 + vv + hi * 8;
      float rowsum = (1.0f - __expf(logg * (float)(i + 1))) * inv1mg;
      float sc = rsqrtf(rowsum);
      ro[(size_t)(rowbase + i) * CV + h * CDV + nt * 16 + ln] = acc[nt][vv] * sc;
    }
}

// ---------------------------------------------------------------------------
// curr_kv[b,h,kk,vv] = sum_j mask[h,T-1,j] * k[b,h,j,kk] * v[b,h,j,vv]
// = (decay-weighted K)^T @ V, one (b,h) per workgroup (256 thr = 8 waves;
// wave w owns kk stripe [w*16, w*16+16), all 256 vv). Both LDS tiles are
// K-pair-interleaved over j so A and B frags are single b64 loads.
// ---------------------------------------------------------------------------
__global__ __launch_bounds__(256) void curr_kv_kernel(
    const float* __restrict__ qkv, float* __restrict__ kvout) {
  __shared__ float kw_s[8][130][2];  // [jpair][kk][0/1]
  __shared__ float v_s[8][258][2];   // [jpair][vv][0/1]

  const int bh = blockIdx.x;
  const int b = bh >> 3, h = bh & 7;
  const int rowbase = b * CT;

  const int tid = threadIdx.x;
  const int lane = tid & 31;
  const int w = tid >> 5;
  const int hi = lane >> 4;
  const int ln = lane & 15;

  const float se = logf(1.0f / 32.0f), ee = logf(1.0f / 512.0f);
  const float g = 1.0f - __expf(se + (ee - se) * ((float)h / 7.0f));
  const float logg = logf(g);
  const float rnorm = rsqrtf((1.0f - __expf(logg * (float)CT)) / (1.0f - g));

  v8f acc[16];
#pragma unroll
  for (int i = 0; i < 16; ++i) acc[i] = vzero8();

  for (int j0 = 0; j0 < CT; j0 += 16) {
#pragma unroll
    for (int e = tid; e < 16 * 128; e += 256) {
      int r = e >> 7, c = e & 127;
      int j = j0 + r;
      float wj = __expf(logg * (float)(CT - 1 - j)) * rnorm;
      kw_s[r >> 1][c][r & 1] =
          qkv[(size_t)(rowbase + j) * NQKV + CE + h * CDK + c] * wj;
    }
#pragma unroll
    for (int e = tid; e < 16 * 256; e += 256) {
      int r = e >> 8, c = e & 255;
      v_s[r >> 1][c][r & 1] =
          qkv[(size_t)(rowbase + j0 + r) * NQKV + 2 * CE + h * CDV + c];
    }
    __syncthreads();

#pragma unroll
    for (int ks = 0; ks < 16; ks += 4) {
      const int kp = (ks >> 1) + hi;
      // A[m=kk][k=j] = kw[j][kk]  (pair over j is interleaved -> one b64)
      v2f a = ldsv2(&kw_s[kp][w * 16 + ln][0]);
#pragma unroll
      for (int nt = 0; nt < 16; ++nt) {
        v2f bf = ldsv2(&v_s[kp][nt * 16 + ln][0]);
        acc[nt] = wmma4(a, bf, acc[nt]);
      }
    }
    __syncthreads();
  }

#pragma unroll
  for (int nt = 0; nt < 16; ++nt)
#pragma unroll
    for (int vv = 0; vv < 8; ++vv) {
      int kk = w * 16 + vv + hi * 8;
      kvout[(size_t)bh * CDK * CDV + (size_t)kk * CDV + nt * 16 + ln] =
          acc[nt][vv];
    }
}

// ---------------------------------------------------------------------------
// GroupNorm (per (b,t,head) over dv=256) + SiLU gate, fused; writes gated
// in place over the gate-projection buffer.
// Grid: BT blocks x 256 threads; wave w handles group/head w (wave32!).
// ---------------------------------------------------------------------------
__global__ __launch_bounds__(256) void gn_gate_kernel(
    const float* __restrict__ ro, float* __restrict__ gbuf) {
  const int row = blockIdx.x;
  const int w = threadIdx.x >> 5;
  const int lane = threadIdx.x & 31;
  const size_t base = (size_t)row * CV + (size_t)w * CDV;

  float x[8];
  float s = 0.0f, s2 = 0.0f;
#pragma unroll
  for (int i = 0; i < 8; ++i) {
    x[i] = ro[base + i * 32 + lane];
    s += x[i];
    s2 += x[i] * x[i];
  }
#pragma unroll
  for (int off = 16; off > 0; off >>= 1) {
    s += __shfl_xor(s, off, 32);
    s2 += __shfl_xor(s2, off, 32);
  }
  const float mu = s * (1.0f / 256.0f);
  const float var = s2 * (1.0f / 256.0f) - mu * mu;
  const float rstd = rsqrtf(var + 1e-5f);

#pragma unroll
  for (int i = 0; i < 8; ++i) {
    size_t idx = base + i * 32 + lane;
    float nrm = (x[i] - mu) * rstd;
    float gv = gbuf[idx];
    float sil = gv / (1.0f + __expf(-gv));
    gbuf[idx] = sil * nrm;
  }
}

// ---------------------------------------------------------------------------
// Launch: inputs in setup_inputs() order.
// d_out = [proj (B*T*E floats) | curr_kv (B*H*dk*dv floats)]
// Workspace: qkv (BT*4096) | gbuf (BT*2048) | ro (BT*2048)  = 128 MB
// ---------------------------------------------------------------------------
extern "C" void kernel_launch(void* const* d_in, const int* in_sizes, int n_in,
                              void* d_out, int out_size, void* d_ws,
                              size_t ws_size, hipStream_t stream) {
  const float* hs    = (const float*)d_in[0];
  const float* w_qkv = (const float*)d_in[1];
  const float* b_qkv = (const float*)d_in[2];
  const float* w_g   = (const float*)d_in[3];
  const float* b_g   = (const float*)d_in[4];
  const float* w_p   = (const float*)d_in[5];
  const float* b_p   = (const float*)d_in[6];

  float* out  = (float*)d_out;
  float* qkv  = (float*)d_ws;
  float* gbuf = qkv + (size_t)BT * NQKV;
  float* ro   = gbuf + (size_t)BT * CV;

  // 1) qkv = hs @ w_qkv + b_qkv     [4096 x 1024 x 4096]
  gemm_wmma_f32<<<dim3(NQKV / 64, BT / 128), 128, 0, stream>>>(
      hs, w_qkv, b_qkv, qkv, BT, NQKV, CE);

  // 2) xPos rotary on q and k, in place
  xpos_kernel<<<dim3(BT), 512, 0, stream>>>(qkv);

  // 3) gate projection gbuf = hs @ w_g + b_g   [4096 x 1024 x 2048]
  gemm_wmma_f32<<<dim3(CV / 64, BT / 128), 128, 0, stream>>>(
      hs, w_g, b_g, gbuf, BT, CV, CE);

  // 4) masked retention -> ro in [B,T,V] layout
  retention_kernel<<<dim3(CT / 64, CB * CH), 128, 0, stream>>>(qkv, ro);

  // 5) curr_kv -> second output region
  curr_kv_kernel<<<dim3(CB * CH), 256, 0, stream>>>(
      qkv, out + (size_t)BT * CE);

  // 6) groupnorm + silu gate (in place over gbuf)
  gn_gate_kernel<<<dim3(BT), 256, 0, stream>>>(ro, gbuf);

  // 7) proj = gated @ w_p + b_p -> first output region [4096 x 2048 x 1024]
  gemm_wmma_f32<<<dim3(CE / 64, BT / 128), 128, 0, stream>>>(
      gbuf, w_p, b_p, out, BT, CE, CV);
}